// DirGCNConv_57432302682556
// MI455X (gfx1250) — compile-verified
//
#include <hip/hip_runtime.h>
#include <hip/hip_bf16.h>

// DirGCNConv on gfx1250:
//   out = ALPHA*(A_norm @ x @ Wsd^T + b_sd) + (1-ALPHA)*(A_norm^T @ x @ Wds^T + b_ds)
// Restructured (linearity): y_sd = x @ Wsd^T, y_ds = x @ Wds^T computed FIRST with
// V_WMMA_F32_16X16X4_F32 (exact fp32), then one fused edge-scatter with f32 atomics.

#define D_FEAT 96
#define ALPHA_C 0.5f

typedef __attribute__((ext_vector_type(2))) float v2f;
typedef __attribute__((ext_vector_type(8))) float v8f;

// ---------------- init: out[n,j] = ALPHA*b_sd[j] + (1-ALPHA)*b_ds[j] ----------------
__global__ void dirgcn_init_out(float* __restrict__ out,
                                const float* __restrict__ b_sd,
                                const float* __restrict__ b_ds,
                                int total) {
  int i = blockIdx.x * blockDim.x + threadIdx.x;
  if (i < total) {
    int j = i % D_FEAT;
    out[i] = ALPHA_C * b_sd[j] + (1.0f - ALPHA_C) * b_ds[j];
  }
}

// ---------------- zero degree buffers ----------------
__global__ void dirgcn_zero2(float* __restrict__ a, float* __restrict__ b, int n) {
  int i = blockIdx.x * blockDim.x + threadIdx.x;
  if (i < n) { a[i] = 0.0f; b[i] = 0.0f; }
}

// ---------------- degree counting ----------------
__global__ void dirgcn_degree(const int* __restrict__ row, const int* __restrict__ col,
                              float* __restrict__ dout, float* __restrict__ din, int E) {
  int e = blockIdx.x * blockDim.x + threadIdx.x;
  if (e < E) {
    atomicAdd(&dout[row[e]], 1.0f);
    atomicAdd(&din[col[e]], 1.0f);
  }
}

// ---------------- deg -> rsqrt(deg) (0 if deg==0), in place ----------------
__global__ void dirgcn_invsqrt(float* __restrict__ a, float* __restrict__ b, int n) {
  int i = blockIdx.x * blockDim.x + threadIdx.x;
  if (i < n) {
    float u = a[i]; a[i] = (u > 0.0f) ? rsqrtf(u) : 0.0f;
    float v = b[i]; b[i] = (v > 0.0f) ? rsqrtf(v) : 0.0f;
  }
}

// ---------------- WMMA GEMM: y_sd = x @ Wsd^T, y_ds = x @ Wds^T ----------------
// One wave per 16x16 output tile. A = x[m0:m0+16, k:k+4] (16x4 f32, 2 VGPRs),
// B = W^T[k:k+4, n0:n0+16] i.e. B[k,o] = W[o,k] (4x16 f32, 2 VGPRs, same half-split
// layout as A per ISA 7.12.2). C/D: 8 VGPRs, VGPR r -> row m0+r+8*(lane>=16),
// col n0+(lane&15). Both weight matrices share the A operand per K step.
__global__ void dirgcn_gemm_wmma(const float* __restrict__ x,
                                 const float* __restrict__ Wsd,
                                 const float* __restrict__ Wds,
                                 float* __restrict__ ysd,
                                 float* __restrict__ yds,
                                 int mtiles, int ntiles) {
  const int lane = threadIdx.x & 31;
  const int wave = threadIdx.x >> 5;
  const int tile = blockIdx.x * (blockDim.x >> 5) + wave;
  if (tile >= mtiles * ntiles) return;   // whole-wave guard: EXEC all-1s for WMMA
  const int mt = tile / ntiles;
  const int nt = tile - mt * ntiles;
  const int m0 = mt * 16;
  const int n0 = nt * 16;
  const int lm = lane & 15;              // row-in-tile (A) / col-in-tile (B,C)
  const int lh = lane >> 4;              // half select: +2 in K for A/B

  const float* ax  = x   + (size_t)(m0 + lm) * D_FEAT + 2 * lh;
  const float* bsd = Wsd + (size_t)(n0 + lm) * D_FEAT + 2 * lh;
  const float* bds = Wds + (size_t)(n0 + lm) * D_FEAT + 2 * lh;

  v8f c_sd = {};
  v8f c_ds = {};
#pragma unroll
  for (int k = 0; k < D_FEAT; k += 4) {
    v2f a  = *(const v2f*)(ax  + k);
    v2f b0 = *(const v2f*)(bsd + k);
    v2f b1 = *(const v2f*)(bds + k);
    // 8 args: (neg_a, A, neg_b, B, c_mod, C, reuse_a, reuse_b)
    c_sd = __builtin_amdgcn_wmma_f32_16x16x4_f32(false, a, false, b0,
                                                 (short)0, c_sd, false, false);
    c_ds = __builtin_amdgcn_wmma_f32_16x16x4_f32(false, a, false, b1,
                                                 (short)0, c_ds, false, false);
  }

  const int colIdx = n0 + lm;
#pragma unroll
  for (int r = 0; r < 8; ++r) {
    const size_t off = (size_t)(m0 + r + 8 * lh) * D_FEAT + colIdx;
    ysd[off] = c_sd[r];
    yds[off] = c_ds[r];
  }
}

// ---------------- scalar tail for rows not covered by 16-row tiles ----------------
__global__ void dirgcn_gemm_tail(const float* __restrict__ x,
                                 const float* __restrict__ Wsd,
                                 const float* __restrict__ Wds,
                                 float* __restrict__ ysd,
                                 float* __restrict__ yds,
                                 int n_start, int n_nodes) {
  int idx = blockIdx.x * blockDim.x + threadIdx.x;
  int total = (n_nodes - n_start) * D_FEAT;
  if (idx >= total) return;
  int r = n_start + idx / D_FEAT;
  int o = idx % D_FEAT;
  float s0 = 0.0f, s1 = 0.0f;
  for (int k = 0; k < D_FEAT; ++k) {
    float xv = x[(size_t)r * D_FEAT + k];
    s0 += xv * Wsd[(size_t)o * D_FEAT + k];
    s1 += xv * Wds[(size_t)o * D_FEAT + k];
  }
  ysd[(size_t)r * D_FEAT + o] = s0;
  yds[(size_t)r * D_FEAT + o] = s1;
}

// ---------------- fused edge scatter ----------------
// 12 threads per edge, each handles 8 contiguous features (four float4 reads,
// 16 f32 atomics). val recomputed per thread: 2 broadcast L0 hits.
__global__ void dirgcn_scatter(const int* __restrict__ row, const int* __restrict__ col,
                               const float* __restrict__ out_is,
                               const float* __restrict__ in_is,
                               const float* __restrict__ ysd,
                               const float* __restrict__ yds,
                               float* __restrict__ out, int E) {
  const int PER_EDGE = D_FEAT / 8;   // 12
  int idx = blockIdx.x * blockDim.x + threadIdx.x;
  if (idx >= E * PER_EDGE) return;
  int e  = idx / PER_EDGE;
  int j0 = (idx - e * PER_EDGE) * 8;
  int r = row[e];
  int c = col[e];
  float v  = out_is[r] * in_is[c];
  float va = ALPHA_C * v;
  float vb = (1.0f - ALPHA_C) * v;

  const float4* s0 = (const float4*)(ysd + (size_t)c * D_FEAT + j0);
  const float4* s1 = (const float4*)(yds + (size_t)r * D_FEAT + j0);
  float* oA = out + (size_t)r * D_FEAT + j0;   // agg direction: col -> row
  float* oB = out + (size_t)c * D_FEAT + j0;   // agg_t direction: row -> col
#pragma unroll
  for (int q = 0; q < 2; ++q) {
    float4 a = s0[q];
    float4 b = s1[q];
    atomicAdd(oA + 4 * q + 0, va * a.x);
    atomicAdd(oA + 4 * q + 1, va * a.y);
    atomicAdd(oA + 4 * q + 2, va * a.z);
    atomicAdd(oA + 4 * q + 3, va * a.w);
    atomicAdd(oB + 4 * q + 0, vb * b.x);
    atomicAdd(oB + 4 * q + 1, vb * b.y);
    atomicAdd(oB + 4 * q + 2, vb * b.z);
    atomicAdd(oB + 4 * q + 3, vb * b.w);
  }
}

extern "C" void kernel_launch(void* const* d_in, const int* in_sizes, int n_in,
                              void* d_out, int out_size, void* d_ws, size_t ws_size,
                              hipStream_t stream) {
  const float* x    = (const float*)d_in[0];
  const float* W_sd = (const float*)d_in[1];
  const float* b_sd = (const float*)d_in[2];
  const float* W_ds = (const float*)d_in[3];
  const float* b_ds = (const float*)d_in[4];
  const int*   ei   = (const int*)d_in[5];

  const int N = in_sizes[0] / D_FEAT;    // 50000
  const int E = in_sizes[5] / 2;         // 800000
  const int* row = ei;                   // edge_index[0]
  const int* col = ei + E;               // edge_index[1]
  float* out = (float*)d_out;

  // Workspace layout (floats): [deg_out N][deg_in N][y_sd N*D][y_ds N*D] ~ 38.8 MB
  float* ws      = (float*)d_ws;
  float* deg_out = ws;
  float* deg_in  = ws + N;
  float* y_sd    = ws + 2 * (size_t)N;
  float* y_ds    = y_sd + (size_t)N * D_FEAT;

  const int T = 256;

  dirgcn_init_out<<<(N * D_FEAT + T - 1) / T, T, 0, stream>>>(out, b_sd, b_ds, N * D_FEAT);
  dirgcn_zero2<<<(N + T - 1) / T, T, 0, stream>>>(deg_out, deg_in, N);
  dirgcn_degree<<<(E + T - 1) / T, T, 0, stream>>>(row, col, deg_out, deg_in, E);
  dirgcn_invsqrt<<<(N + T - 1) / T, T, 0, stream>>>(deg_out, deg_in, N);

  const int mtiles = N / 16;             // 3125 (exact for N=50000)
  const int ntiles = D_FEAT / 16;        // 6
  const int tiles  = mtiles * ntiles;    // 18750 waves
  const int wpb    = T / 32;             // 8 waves/block
  dirgcn_gemm_wmma<<<(tiles + wpb - 1) / wpb, T, 0, stream>>>(
      x, W_sd, W_ds, y_sd, y_ds, mtiles, ntiles);
  const int rem_start = mtiles * 16;
  if (rem_start < N) {
    dirgcn_gemm_tail<<<((N - rem_start) * D_FEAT + T - 1) / T, T, 0, stream>>>(
        x, W_sd, W_ds, y_sd, y_ds, rem_start, N);
  }

  const int sc_total = E * (D_FEAT / 8); // 9.6M threads
  dirgcn_scatter<<<(sc_total + T - 1) / T, T, 0, stream>>>(
      row, col, deg_out, deg_in, y_sd, y_ds, out, E);
}